// GeometryAwareAttention_43791486550414
// MI455X (gfx1250) — compile-verified
//
#include <hip/hip_runtime.h>
#include <hip/hip_bf16.h>
#include <math.h>
#include <stdint.h>

#define NN  2048
#define DD  256
#define HH  8
#define MMM 4
#define DKK 32
#define EPSF 1e-6f
#define JSPLIT 4                    // j split -> 512 attention blocks
#define JTILES (NN / (16 * JSPLIT)) // 32 j-tiles per block

typedef float v2f __attribute__((ext_vector_type(2)));
typedef float v8f __attribute__((ext_vector_type(8)));

// D = A(16x4 f32) * B(4x16 f32) + C ; exact f32 semantics (matches reference)
__device__ __forceinline__ v8f wmma4(v2f a, v2f b, v8f c) {
  return __builtin_amdgcn_wmma_f32_16x16x4_f32(false, a, false, b, (short)0, c,
                                               false, false);
}

__device__ __forceinline__ v2f ld2(const float* p) { return *(const v2f*)p; }

// ---------------------------------------------------------------------------
// GEMM: C[i][j] = sum_d A[i][d] * W[j][d]   (A:[NN x DD], W:[DD x DD])
// One wave -> 16 rows x 32 cols (2 accumulators); 1024 waves = 128 blocks.
// A layout (ISA 7.12.2, 32-bit A 16x4): lane m = L%16, k = 2*(L>=16)+v
// B layout (rows interleaved like C/D):  lane n = L%16, k = 2*(L>=16)+v
// ---------------------------------------------------------------------------
__device__ __forceinline__ void gemm_xwt(const float* __restrict__ A,
                                         const float* __restrict__ W,
                                         float* __restrict__ C) {
  const int w    = threadIdx.x >> 5;
  const int lane = threadIdx.x & 31;
  const int n16  = lane & 15;
  const int hi   = lane >> 4;
  const int gw   = blockIdx.x * 8 + w;   // 1024 waves total
  const int i0   = (gw >> 3) * 16;       // 128 row tiles
  const int j0   = (gw & 7) * 32;        // 8 col tiles of 32

  v8f acc[2] = {};
  const float* arow = A + (size_t)(i0 + n16) * DD + 2 * hi;
#pragma unroll 4
  for (int d0 = 0; d0 < DD; d0 += 4) {
    v2f a = ld2(arow + d0);
#pragma unroll
    for (int c = 0; c < 2; ++c) {
      v2f b = ld2(W + (size_t)(j0 + 16 * c + n16) * DD + d0 + 2 * hi);
      acc[c] = wmma4(a, b, acc[c]);
    }
  }
#pragma unroll
  for (int c = 0; c < 2; ++c)
#pragma unroll
    for (int r = 0; r < 8; ++r)
      C[(size_t)(i0 + r + 8 * hi) * DD + j0 + 16 * c + n16] = acc[c][r];
}

__global__ __launch_bounds__(256) void qkv_kernel(
    const float* __restrict__ x, const float* __restrict__ wq,
    const float* __restrict__ wk, const float* __restrict__ wv,
    float* __restrict__ Q, float* __restrict__ K, float* __restrict__ V) {
  const int z = blockIdx.z;
  const float* W = (z == 0) ? wq : (z == 1) ? wk : wv;
  float* C       = (z == 0) ? Q  : (z == 1) ? K  : V;
  gemm_xwt(x, W, C);
}

__global__ __launch_bounds__(256) void oproj_kernel(
    const float* __restrict__ A, const float* __restrict__ wo,
    float* __restrict__ C) {
  gemm_xwt(A, wo, C);
}

// ---------------------------------------------------------------------------
// Fused flash attention with graph bias.
// grid = (128 i-tiles, JSPLIT j-chunks); block = 256 thr = 8 waves = 8 heads.
// adjs tiles are double-buffered in LDS and fetched one tile ahead with
// GLOBAL_LOAD_ASYNC_TO_LDS_B128 (ASYNCcnt), overlapping DMA with WMMA+softmax.
// ---------------------------------------------------------------------------
__global__ __launch_bounds__(256) void attn_kernel(
    const float* __restrict__ Q, const float* __restrict__ Km,
    const float* __restrict__ V, const float* __restrict__ pi,
    const float* __restrict__ tau, const float* __restrict__ adjs,
    float* __restrict__ Opart, float* __restrict__ Mp, float* __restrict__ Lp) {
  __shared__ __align__(16) float s_adj[2][MMM][16][20];  // double buffer, padded
  __shared__ float s_p[HH][16][17];                      // per-wave P transpose

  const int tid  = threadIdx.x;
  const int h    = tid >> 5;            // wave == head
  const int lane = tid & 31;
  const int n16  = lane & 15;
  const int hi   = lane >> 4;
  const int i0   = blockIdx.x * 16;
  const int jc   = blockIdx.y;
  const int jbase = jc * (NN / JSPLIT);
  const float tau_h = tau[h];
  const float scale = 0.17677669529663688f;  // 1/sqrt(DK)

  // Async-staging geometry: one b128 per thread covers a 4KB tile.
  const int am  = tid >> 6;
  const int ail = (tid >> 2) & 15;
  const int aj4 = (tid & 3) * 4;
  const uint32_t lds_dst0 = (uint32_t)(uintptr_t)&s_adj[0][am][ail][aj4];
  const uint32_t lds_dst1 = (uint32_t)(uintptr_t)&s_adj[1][am][ail][aj4];
  const size_t adj_row = ((size_t)am * NN + (i0 + ail)) * NN;

#define ISSUE_ADJ(JT, BUF)                                                    \
  do {                                                                        \
    const uint32_t goff =                                                     \
        (uint32_t)((adj_row + (jbase + (JT) * 16) + aj4) * sizeof(float));    \
    asm volatile("global_load_async_to_lds_b128 %0, %1, %2" ::"v"(            \
                     (BUF) ? lds_dst1 : lds_dst0),                            \
                 "v"(goff), "s"(adjs)                                         \
                 : "memory");                                                 \
  } while (0)

  // Preload Q rows (scale folded in).  a[v]: m=n16, k=2*hi+v within K-chunk.
  v2f qa[8];
  const float* qrow = Q + (size_t)(i0 + n16) * DD + h * DKK + 2 * hi;
#pragma unroll
  for (int kk = 0; kk < 8; ++kk) {
    v2f t = ld2(qrow + kk * 4);
    t.x *= scale; t.y *= scale;
    qa[kk] = t;
  }
  // tau * pi for the 8 rows this lane owns in C-layout (row = r + 8*hi).
  float tpi[8][MMM];
#pragma unroll
  for (int r = 0; r < 8; ++r) {
    const float* pr = pi + (size_t)(i0 + r + 8 * hi) * (HH * MMM) + h * MMM;
#pragma unroll
    for (int m = 0; m < MMM; ++m) tpi[r][m] = tau_h * pr[m];
  }

  v8f o0 = {}, o1 = {};
  float mrow[8], lrow[8];
#pragma unroll
  for (int r = 0; r < 8; ++r) { mrow[r] = -INFINITY; lrow[r] = 0.0f; }

  ISSUE_ADJ(0, 0);  // prologue: DMA first tile

  for (int jt = 0; jt < JTILES; ++jt) {
    const int j0  = jbase + jt * 16;
    const int cur = jt & 1;

    if (jt + 1 < JTILES) {
      ISSUE_ADJ(jt + 1, cur ^ 1);                        // run one tile ahead
      asm volatile("s_wait_asynccnt 0x1" ::: "memory");  // cur tile landed
    } else {
      asm volatile("s_wait_asynccnt 0x0" ::: "memory");  // drain on last tile
    }
    __syncthreads();  // every wave's wait done -> cur buffer valid block-wide

    // scores S = Q * K^T  (K=32 in 8 steps of 4)
    v8f s = {};
    const float* krow = Km + (size_t)(j0 + n16) * DD + h * DKK + 2 * hi;
#pragma unroll
    for (int kk = 0; kk < 8; ++kk) s = wmma4(qa[kk], ld2(krow + kk * 4), s);

    // bias + online softmax in C-layout (row = r + 8*hi, col = n16)
#pragma unroll
    for (int r = 0; r < 8; ++r) {
      float aeff = EPSF;
#pragma unroll
      for (int m = 0; m < MMM; ++m)
        aeff = fmaf(tpi[r][m], s_adj[cur][m][r + 8 * hi][n16], aeff);
      float xv = s[r] + __logf(aeff);
      float rmax = xv;
#pragma unroll
      for (int off = 1; off < 16; off <<= 1)
        rmax = fmaxf(rmax, __shfl_xor(rmax, off, 16));
      const float mnew  = fmaxf(mrow[r], rmax);
      const float p     = __expf(xv - mnew);
      const float alpha = __expf(mrow[r] - mnew);
      float rsum = p;
#pragma unroll
      for (int off = 1; off < 16; off <<= 1) rsum += __shfl_xor(rsum, off, 16);
      lrow[r] = lrow[r] * alpha + rsum;
      mrow[r] = mnew;
      o0[r] *= alpha; o1[r] *= alpha;
      s_p[h][r + 8 * hi][n16] = p;   // stash P for A-layout reload
    }
    __builtin_amdgcn_wave_barrier();
    asm volatile("s_wait_dscnt 0x0" ::: "memory");  // P stores visible in-wave

    // O += P(16x16) * V_tile(16x32): A from LDS, B from V, 2 N-tiles x 4 K.
#pragma unroll
    for (int kk = 0; kk < 4; ++kk) {
      v2f a;
      a.x = s_p[h][n16][kk * 4 + 2 * hi];
      a.y = s_p[h][n16][kk * 4 + 2 * hi + 1];
      const float* v0 = V + (size_t)(j0 + kk * 4 + 2 * hi) * DD + h * DKK + n16;
      const float* v1 = v0 + DD;
      v2f b0; b0.x = v0[0];  b0.y = v1[0];
      v2f b1; b1.x = v0[16]; b1.y = v1[16];
      o0 = wmma4(a, b0, o0);
      o1 = wmma4(a, b1, o1);
    }
    __syncthreads();  // all reads of cur done before it is re-targeted (jt+2)
  }
#undef ISSUE_ADJ

  // Store unnormalized partials + per-row flash stats for this j-chunk.
  float* op = Opart + (size_t)jc * NN * DD;
#pragma unroll
  for (int r = 0; r < 8; ++r) {
    float* orow = op + (size_t)(i0 + r + 8 * hi) * DD + h * DKK;
    orow[n16]      = o0[r];
    orow[16 + n16] = o1[r];
  }
  if (n16 == 0) {
#pragma unroll
    for (int r = 0; r < 8; ++r) {
      const size_t si = ((size_t)jc * NN + (i0 + r + 8 * hi)) * HH + h;
      Mp[si] = mrow[r];
      Lp[si] = lrow[r];
    }
  }
}

// Merge the JSPLIT flash partials: OH = sum_c O_c*exp(m_c-m*) / sum_c l_c*exp(m_c-m*)
__global__ __launch_bounds__(256) void combine_kernel(
    const float* __restrict__ Opart, const float* __restrict__ Mp,
    const float* __restrict__ Lp, float* __restrict__ OH) {
  const int idx = blockIdx.x * 256 + threadIdx.x;  // over NN*DD
  const int i = idx >> 8;
  const int h = (idx & 255) >> 5;
  float mstar = -INFINITY;
#pragma unroll
  for (int c = 0; c < JSPLIT; ++c)
    mstar = fmaxf(mstar, Mp[((size_t)c * NN + i) * HH + h]);
  float lstar = 0.0f, acc = 0.0f;
#pragma unroll
  for (int c = 0; c < JSPLIT; ++c) {
    const size_t si = ((size_t)c * NN + i) * HH + h;
    const float wgt = __expf(Mp[si] - mstar);
    lstar = fmaf(Lp[si], wgt, lstar);
    acc   = fmaf(Opart[(size_t)c * NN * DD + idx], wgt, acc);
  }
  OH[idx] = acc / lstar;
}

extern "C" void kernel_launch(void* const* d_in, const int* in_sizes, int n_in,
                              void* d_out, int out_size, void* d_ws,
                              size_t ws_size, hipStream_t stream) {
  const float* x    = (const float*)d_in[0];
  const float* pi   = (const float*)d_in[1];
  const float* tau  = (const float*)d_in[2];
  const float* adjs = (const float*)d_in[3];
  const float* Wq   = (const float*)d_in[4];
  const float* Wk   = (const float*)d_in[5];
  const float* Wv   = (const float*)d_in[6];
  const float* Wo   = (const float*)d_in[7];
  float* out = (float*)d_out;

  float* ws = (float*)d_ws;
  const size_t ND = (size_t)NN * DD;      // 524288
  float* Q     = ws;                      ws += ND;
  float* K     = ws;                      ws += ND;
  float* V     = ws;                      ws += ND;
  float* Opart = ws;                      ws += ND * JSPLIT;
  float* Mp    = ws;                      ws += (size_t)JSPLIT * NN * HH;
  float* Lp    = ws;                      ws += (size_t)JSPLIT * NN * HH;
  float* OH    = ws;                      ws += ND;

  qkv_kernel<<<dim3(128, 1, 3), 256, 0, stream>>>(x, Wq, Wk, Wv, Q, K, V);
  attn_kernel<<<dim3(NN / 16, JSPLIT), 256, 0, stream>>>(Q, K, V, pi, tau,
                                                         adjs, Opart, Mp, Lp);
  combine_kernel<<<dim3((unsigned)(ND / 256)), 256, 0, stream>>>(Opart, Mp, Lp,
                                                                 OH);
  oproj_kernel<<<dim3(128), 256, 0, stream>>>(OH, Wo, out);
}